// AttentionCell_5222680232089
// MI455X (gfx1250) — compile-verified
//
#include <hip/hip_runtime.h>
#include <math.h>

// Problem constants (hardcoded in reference)
#define NB   512   // batch
#define CC   512   // channels
#define EE   128   // embedding
#define HWC  240   // H*W = 6*40
#define XK   640   // C+E (K for gi GEMM)
#define G3   1536  // 3C

typedef float v2f __attribute__((ext_vector_type(2)));
typedef float v8f __attribute__((ext_vector_type(8)));

// ---------------------------------------------------------------------------
// Kernel 1: attention scores + softmax + context. One block per batch item.
// (bandwidth-bound part: streams conv_feats / conv_feats_origin once, 502 MB
//  total across the grid -> ~21.5 us floor at 23.3 TB/s)
// ---------------------------------------------------------------------------
__global__ __launch_bounds__(256)
void attn_context_kernel(const float* __restrict__ prev_h,   // [B,C]
                         const float* __restrict__ cf,       // [B,C,HW]
                         const float* __restrict__ cfo,      // [B,C,HW]
                         const float* __restrict__ emb,      // [B,E]
                         const float* __restrict__ score_w,  // [C]
                         float* __restrict__ x_buf,          // [B,XK]
                         float* __restrict__ alphaT)         // [HW,B]
{
  __shared__ float sh_h[CC];
  __shared__ float sh_w[CC];
  __shared__ float sh_a[HWC];
  __shared__ float red[256];

  const int b = blockIdx.x;
  const int t = threadIdx.x;

  for (int c = t; c < CC; c += 256) {
    sh_h[c] = prev_h[b * CC + c];
    sh_w[c] = score_w[c];
  }
  __syncthreads();

  // ---- additive attention score for position t ----
  float em = -INFINITY;
  float score = 0.0f;
  if (t < HWC) {
    const float* p = cf + (size_t)b * CC * HWC + t;
    #pragma unroll 4
    for (int c = 0; c < CC; ++c)
      score += tanhf(p[(size_t)c * HWC] + sh_h[c]) * sh_w[c];
    em = score;
  }

  // ---- softmax over HW (block reduction) ----
  red[t] = em;
  __syncthreads();
  for (int s = 128; s > 0; s >>= 1) {
    if (t < s) red[t] = fmaxf(red[t], red[t + s]);
    __syncthreads();
  }
  const float mx = red[0];
  __syncthreads();

  float ex = 0.0f;
  if (t < HWC) ex = expf(score - mx);
  red[t] = ex;
  __syncthreads();
  for (int s = 128; s > 0; s >>= 1) {
    if (t < s) red[t] += red[t + s];
    __syncthreads();
  }
  const float inv = 1.0f / red[0];
  if (t < HWC) {
    const float a = ex * inv;
    sh_a[t] = a;
    alphaT[(size_t)t * NB + b] = a;   // alpha returned transposed [HW,B]
  }
  __syncthreads();

  // ---- context[b,c] = sum_h cfo[b,c,h] * alpha[b,h] ----
  const int lane = t & 31;
  const int wave = t >> 5;
  for (int c = wave; c < CC; c += 8) {
    const float* q = cfo + (size_t)b * CC * HWC + (size_t)c * HWC;
    float s = 0.0f;
    for (int h = lane; h < HWC; h += 32)
      s += q[h] * sh_a[h];
    #pragma unroll
    for (int off = 16; off > 0; off >>= 1)
      s += __shfl_xor(s, off, 32);
    if (lane == 0) x_buf[(size_t)b * XK + c] = s;
  }

  // ---- x[:, C:] = embeddings ----
  for (int e = t; e < EE; e += 256)
    x_buf[(size_t)b * XK + CC + e] = emb[b * EE + e];
}

// ---------------------------------------------------------------------------
// Kernel 2: D[M,N] = A[M,K] * W[N,K]^T + bias via V_WMMA_F32_16X16X4_F32.
//
// Register-tiled: each wave computes a 32(M) x 64(N) macro-tile with
// 2x4 = 8 v8f accumulators. Per k-step of 4: 6 b64 loads feed 8 WMMAs
// (A reused 4x, B reused 2x). Block = 256 threads = 8 waves laid out
// along N (block tile = 32M x 512N); all reuse is in registers, no LDS.
//
// VGPR layouts per CDNA5 ISA 7.12.2:
//   A (16x4, f32, 2 VGPRs): lane&15 = M row, (lane>>4)*2 = K pair base
//   B (4x16, f32, 2 VGPRs): lane&15 = N col, (lane>>4)*2 = K pair base
//   C/D (16x16, 8 VGPRs):   row = r + (lane>>4)*8, col = lane&15
// Since B = W^T and W is [N,K] row-major, B loads use the same
// row*K + k addressing as A.
// ---------------------------------------------------------------------------
__global__ __launch_bounds__(256)
void gemm_wmma_f32_tiled(const float* __restrict__ A,    // [M,K]
                         const float* __restrict__ W,    // [N,K]
                         const float* __restrict__ bias, // [N]
                         float* __restrict__ D,          // [M,G3]
                         int K)
{
  const int lane = threadIdx.x & 31;
  const int wave = threadIdx.x >> 5;
  const int m0 = blockIdx.x * 32;
  const int n0 = blockIdx.y * 512 + wave * 64;
  const int mn = lane & 15;
  const int kk = (lane >> 4) * 2;

  const float* pa0 = A + (size_t)(m0 +      mn) * K + kk;
  const float* pa1 = A + (size_t)(m0 + 16 + mn) * K + kk;
  const float* pb  = W + (size_t)(n0 +      mn) * K + kk;
  const size_t bstride = (size_t)16 * K;   // between N sub-tiles

  v8f acc[2][4] = {};

  for (int k = 0; k < K; k += 4) {
    v2f a0, a1, bv[4];
    a0.x = pa0[k];  a0.y = pa0[k + 1];
    a1.x = pa1[k];  a1.y = pa1[k + 1];
    #pragma unroll
    for (int j = 0; j < 4; ++j) {
      bv[j].x = pb[(size_t)j * bstride + k];
      bv[j].y = pb[(size_t)j * bstride + k + 1];
    }
    #pragma unroll
    for (int j = 0; j < 4; ++j) {
      acc[0][j] = __builtin_amdgcn_wmma_f32_16x16x4_f32(
          false, a0, false, bv[j], (short)0, acc[0][j], false, false);
      acc[1][j] = __builtin_amdgcn_wmma_f32_16x16x4_f32(
          false, a1, false, bv[j], (short)0, acc[1][j], false, false);
    }
  }

  // ---- store with fused bias ----
  const int mrow = (lane >> 4) * 8;   // D row offset for upper half-wave
  #pragma unroll
  for (int j = 0; j < 4; ++j) {
    const float bs = bias[n0 + 16 * j + mn];
    #pragma unroll
    for (int i = 0; i < 2; ++i) {
      float* pd = D + (size_t)(m0 + 16 * i + mrow) * G3 + n0 + 16 * j + mn;
      #pragma unroll
      for (int r = 0; r < 8; ++r)
        pd[(size_t)r * G3] = acc[i][j][r] + bs;
    }
  }
}

// ---------------------------------------------------------------------------
// Kernel 3: GRU gate math (PyTorch GRUCell semantics).
// ---------------------------------------------------------------------------
__global__ __launch_bounds__(256)
void gru_gate_kernel(const float* __restrict__ gi,    // [B,3C]
                     const float* __restrict__ gh,    // [B,3C]
                     const float* __restrict__ hprev, // [B,C]
                     float* __restrict__ out)         // [B,C]
{
  const int idx = blockIdx.x * 256 + threadIdx.x;
  if (idx >= NB * CC) return;
  const int b = idx >> 9;      // / 512
  const int j = idx & (CC - 1);
  const float* pi = gi + (size_t)b * G3;
  const float* ph = gh + (size_t)b * G3;

  const float r = 1.0f / (1.0f + expf(-(pi[j] + ph[j])));
  const float z = 1.0f / (1.0f + expf(-(pi[CC + j] + ph[CC + j])));
  const float n = tanhf(pi[2 * CC + j] + r * ph[2 * CC + j]);
  out[idx] = (1.0f - z) * n + z * hprev[idx];
}

// ---------------------------------------------------------------------------
extern "C" void kernel_launch(void* const* d_in, const int* in_sizes, int n_in,
                              void* d_out, int out_size, void* d_ws, size_t ws_size,
                              hipStream_t stream) {
  (void)in_sizes; (void)n_in; (void)out_size; (void)ws_size;

  const float* prev_hidden = (const float*)d_in[0];  // [B,C]
  const float* conv_feats  = (const float*)d_in[1];  // [B,C,H,W]
  const float* conv_feats_o= (const float*)d_in[2];  // [B,C,H,W]
  const float* cur_emb     = (const float*)d_in[3];  // [B,E]
  const float* score_w     = (const float*)d_in[4];  // [C]
  const float* w_ih        = (const float*)d_in[5];  // [3C, C+E]
  const float* w_hh        = (const float*)d_in[6];  // [3C, C]
  const float* b_ih        = (const float*)d_in[7];  // [3C]
  const float* b_hh        = (const float*)d_in[8];  // [3C]

  float* out = (float*)d_out;
  float* out_hidden = out;            // [B,C]
  float* out_alphaT = out + NB * CC;  // [HW,B]

  float* ws    = (float*)d_ws;
  float* x_buf = ws;                       // [B, XK]   = 327680 f
  float* gi    = x_buf + (size_t)NB * XK;  // [B, 3C]   = 786432 f
  float* gh    = gi    + (size_t)NB * G3;  // [B, 3C]   = 786432 f

  // 1) attention + softmax + context -> x_buf, alpha^T
  attn_context_kernel<<<NB, 256, 0, stream>>>(
      prev_hidden, conv_feats, conv_feats_o, cur_emb, score_w, x_buf, out_alphaT);

  // 2) gi = x @ w_ih^T + b_ih   (M=512, N=1536, K=640)
  gemm_wmma_f32_tiled<<<dim3(NB / 32, G3 / 512), 256, 0, stream>>>(
      x_buf, w_ih, b_ih, gi, XK);

  //    gh = h @ w_hh^T + b_hh   (M=512, N=1536, K=512)
  gemm_wmma_f32_tiled<<<dim3(NB / 32, G3 / 512), 256, 0, stream>>>(
      prev_hidden, w_hh, b_hh, gh, CC);

  // 3) GRU gates -> cur_hidden
  gru_gate_kernel<<<(NB * CC + 255) / 256, 256, 0, stream>>>(
      gi, gh, prev_hidden, out_hidden);
}